// ResFeature_85031762526730
// MI455X (gfx1250) — compile-verified
//
#include <hip/hip_runtime.h>
#include <hip/hip_bf16.h>
#include <math.h>

#define BB   16
#define LL   2048
#define AT   6
#define KK   32
#define NRES (BB * LL)        // 32768 residues
#define EPSF 1e-6f

typedef __attribute__((ext_vector_type(2))) float v2f;
typedef __attribute__((ext_vector_type(8))) float v8f;

__device__ __forceinline__ float gelu_exact(float x) {
    return 0.5f * x * (1.0f + erff(x * 0.70710678118654752440f));
}

__device__ __forceinline__ float dot3p(const float* a, const float* b) {
    return a[0] * b[0] + a[1] * b[1] + a[2] * b[2];
}

__device__ __forceinline__ void cross3(float* o, const float* a, const float* b) {
    o[0] = a[1] * b[2] - a[2] * b[1];
    o[1] = a[2] * b[0] - a[0] * b[2];
    o[2] = a[0] * b[1] - a[1] * b[0];
}

// D = A(16x4,f32) * B(4x16,f32) + C(16x16,f32) on one wave32.
__device__ __forceinline__ v8f wmma4(v2f a, v2f b, v8f c) {
    return __builtin_amdgcn_wmma_f32_16x16x4_f32(false, a, false, b, (short)0, c,
                                                 false, false);
}

// ---------------------------------------------------------------------------
// Kernel 1: per-residue scalar features.
//   raw[10] = 5 bond lengths, 3 angle-cos, 2 dihedral-cos (masked)
//   pooling MLP (10->32->6) + softmax over atoms + pooled atom embedding (32)
//   ws_h[res][44]  = [raw(10) | pooled(32) | 0, 0]      (padded for K=44 WMMA)
//   ws_geo[res][21]= [bin(4x3) normalized bonds | nin(3x3) normalized normals]
//   ws_avg[res][3] = atom-mean coords
// ---------------------------------------------------------------------------
__global__ __launch_bounds__(256) void k_residue(
    const float* __restrict__ coords, const float* __restrict__ mask,
    const float* __restrict__ atom_emb, const float* __restrict__ atom_mask,
    const float* __restrict__ pw1, const float* __restrict__ pb1,
    const float* __restrict__ pw2, const float* __restrict__ pb2,
    float* __restrict__ ws_avg, float* __restrict__ ws_h,
    float* __restrict__ ws_geo)
{
    int r = blockIdx.x * blockDim.x + threadIdx.x;
    if (r >= NRES) return;

    const float* c = coords + (size_t)r * (AT * 3);
    float cl[AT][3];
#pragma unroll
    for (int a = 0; a < AT; ++a)
#pragma unroll
        for (int d = 0; d < 3; ++d) cl[a][d] = c[a * 3 + d];

    // mean over atoms
#pragma unroll
    for (int d = 0; d < 3; ++d) {
        float s = 0.f;
#pragma unroll
        for (int a = 0; a < AT; ++a) s += cl[a][d];
        ws_avg[(size_t)r * 3 + d] = s * (1.0f / AT);
    }

    // bonds
    float bo[5][3];
#pragma unroll
    for (int a = 0; a < 5; ++a)
#pragma unroll
        for (int d = 0; d < 3; ++d) bo[a][d] = cl[a + 1][d] - cl[a][d];

    float raw[10];
#pragma unroll
    for (int a = 0; a < 5; ++a) raw[a] = sqrtf(dot3p(bo[a], bo[a]) + EPSF);
#pragma unroll
    for (int a = 0; a < 3; ++a) {
        float n1 = sqrtf(dot3p(bo[a], bo[a]));
        float n2 = sqrtf(dot3p(bo[a + 1], bo[a + 1]));
        raw[5 + a] = (-dot3p(bo[a], bo[a + 1])) / (n1 * n2 + EPSF);
    }
    float nn[3][3];
#pragma unroll
    for (int a = 0; a < 3; ++a) cross3(nn[a], bo[a], bo[a + 1]);
#pragma unroll
    for (int a = 0; a < 2; ++a) {
        float n1 = sqrtf(dot3p(nn[a], nn[a]));
        float n2 = sqrtf(dot3p(nn[a + 1], nn[a + 1]));
        raw[8 + a] = dot3p(nn[a], nn[a + 1]) / (n1 * n2 + EPSF);
    }
    float mi = mask[r];
#pragma unroll
    for (int t = 0; t < 10; ++t) raw[t] *= mi;

    // pooling MLP: gelu(raw @ pw1 + pb1) @ pw2 + pb2  -> logits over 6 atoms
    float w6[AT];
    {
        float hh[32];
#pragma unroll 4
        for (int o = 0; o < 32; ++o) {
            float s = pb1[o];
#pragma unroll
            for (int i = 0; i < 10; ++i) s += raw[i] * pw1[i * 32 + o];
            hh[o] = gelu_exact(s);
        }
#pragma unroll
        for (int o = 0; o < AT; ++o) {
            float s = pb2[o];
#pragma unroll 4
            for (int i = 0; i < 32; ++i) s += hh[i] * pw2[i * AT + o];
            w6[o] = s;
        }
    }
    const float* am = atom_mask + (size_t)r * AT;
#pragma unroll
    for (int o = 0; o < AT; ++o) w6[o] *= am[o];
    float mx = w6[0];
#pragma unroll
    for (int o = 1; o < AT; ++o) mx = fmaxf(mx, w6[o]);
    float ssum = 0.f;
#pragma unroll
    for (int o = 0; o < AT; ++o) { w6[o] = expf(w6[o] - mx); ssum += w6[o]; }
    float inv = 1.0f / ssum;
#pragma unroll
    for (int o = 0; o < AT; ++o) w6[o] *= inv;

    const float* ae = atom_emb + (size_t)r * AT * 32;
    float* hp = ws_h + (size_t)r * 44;
#pragma unroll
    for (int t = 0; t < 10; ++t) hp[t] = raw[t];
#pragma unroll 4
    for (int d2 = 0; d2 < 32; ++d2) {
        float s = 0.f;
#pragma unroll
        for (int a = 0; a < AT; ++a) s += ae[a * 32 + d2] * w6[a];
        hp[10 + d2] = s;
    }
    hp[42] = 0.f;
    hp[43] = 0.f;

    // normalized bonds / normals for the edge kernel
    float* gp = ws_geo + (size_t)r * 21;
#pragma unroll
    for (int a = 0; a < 4; ++a) {
        float nv = sqrtf(dot3p(bo[a], bo[a])) + EPSF;
#pragma unroll
        for (int d = 0; d < 3; ++d) gp[a * 3 + d] = bo[a][d] / nv;
    }
#pragma unroll
    for (int a = 0; a < 3; ++a) {
        float nv = sqrtf(dot3p(nn[a], nn[a])) + EPSF;
#pragma unroll
        for (int d = 0; d < 3; ++d) gp[12 + a * 3 + d] = nn[a][d] / nv;
    }
}

// ---------------------------------------------------------------------------
// Kernel 2: top-K (K=32) nearest neighbors per residue. One wave32 per row.
// Iterative argmin extraction with (dist,index) packed 64-bit keys so ties
// pick the lowest index first, matching lax.top_k ordering on -dist.
// ---------------------------------------------------------------------------
__global__ __launch_bounds__(32) void k_topk(
    const float* __restrict__ ws_avg, const float* __restrict__ mask,
    int* __restrict__ eidx, float* __restrict__ eidx_f)
{
    __shared__ float dist[LL];
    int row = blockIdx.x;
    int lane = threadIdx.x;
    int b = row >> 11;            // row / LL
    int base = b << 11;           // b * LL
    float ax = ws_avg[(size_t)row * 3 + 0];
    float ay = ws_avg[(size_t)row * 3 + 1];
    float az = ws_avg[(size_t)row * 3 + 2];
    float mi = mask[row];

    for (int j = lane; j < LL; j += 32) {
        float dx = ax - ws_avg[(size_t)(base + j) * 3 + 0];
        float dy = ay - ws_avg[(size_t)(base + j) * 3 + 1];
        float dz = az - ws_avg[(size_t)(base + j) * 3 + 2];
        float d = sqrtf(dx * dx + dy * dy + dz * dz + EPSF);
        float m2 = mi * mask[base + j];
        dist[j] = d * m2 + (1.0f - m2) * 1000000.0f;
    }
    __syncthreads();

    for (int k = 0; k < KK; ++k) {
        unsigned long long best = ~0ull;
        for (int j = lane; j < LL; j += 32) {
            unsigned long long key =
                ((unsigned long long)__float_as_uint(dist[j]) << 32) | (unsigned)j;
            best = key < best ? key : best;
        }
#pragma unroll
        for (int off = 16; off > 0; off >>= 1) {
            unsigned long long o = __shfl_xor(best, off, 32);
            best = o < best ? o : best;
        }
        int widx = (int)(best & 0xffffffffull);
        if (lane == 0) {
            int v = (mi > 0.f) ? widx : -1;
            eidx[(size_t)row * KK + k] = v;
            eidx_f[(size_t)row * KK + k] = (float)v;
            dist[widx] = __uint_as_float(0x7f800000u);  // +inf: exclude
        }
        __syncthreads();
    }
}

// ---------------------------------------------------------------------------
// Kernel 3: residue MLP via f32 WMMA. Each wave computes one 16-residue tile:
//   res_emb = gelu(gelu(h(42,pad44) @ rw1 + rb1) @ rw2 + rb2)
// ---------------------------------------------------------------------------
__global__ __launch_bounds__(256) void k_resmlp(
    const float* __restrict__ ws_h,
    const float* __restrict__ rw1, const float* __restrict__ rb1,
    const float* __restrict__ rw2, const float* __restrict__ rb2,
    float* __restrict__ res_out)
{
    __shared__ float Hs[8][16][32];
    int lane = threadIdx.x & 31;
    int wv = threadIdx.x >> 5;
    int tile = blockIdx.x * 8 + wv;   // 2048 tiles total, exact
    int l15 = lane & 15;
    int koff = (lane < 16) ? 0 : 2;
    int rowoff = (lane < 16) ? 0 : 8;
    const float* hb = ws_h + (size_t)(tile * 16) * 44;

    v8f acc0 = {0.f, 0.f, 0.f, 0.f, 0.f, 0.f, 0.f, 0.f};
    v8f acc1 = acc0;
#pragma unroll
    for (int c = 0; c < 11; ++c) {          // K = 44 (rows 42,43 are zero)
        int kk = 4 * c + koff;
        v2f a;
        a.x = hb[l15 * 44 + kk];
        a.y = hb[l15 * 44 + kk + 1];
        v2f b0, b1;
        b0.x = (kk < 42) ? rw1[kk * 32 + l15] : 0.f;
        b0.y = (kk + 1 < 42) ? rw1[(kk + 1) * 32 + l15] : 0.f;
        b1.x = (kk < 42) ? rw1[kk * 32 + 16 + l15] : 0.f;
        b1.y = (kk + 1 < 42) ? rw1[(kk + 1) * 32 + 16 + l15] : 0.f;
        acc0 = wmma4(a, b0, acc0);
        acc1 = wmma4(a, b1, acc1);
    }
    float bias0 = rb1[l15], bias1 = rb1[16 + l15];
#pragma unroll
    for (int rr = 0; rr < 8; ++rr) {
        Hs[wv][rr + rowoff][l15]      = gelu_exact(acc0[rr] + bias0);
        Hs[wv][rr + rowoff][16 + l15] = gelu_exact(acc1[rr] + bias1);
    }
    __syncthreads();

    v8f acc2 = {0.f, 0.f, 0.f, 0.f, 0.f, 0.f, 0.f, 0.f};
    v8f acc3 = acc2;
#pragma unroll
    for (int c = 0; c < 8; ++c) {           // K = 32
        int kk = 4 * c + koff;
        v2f a;
        a.x = Hs[wv][l15][kk];
        a.y = Hs[wv][l15][kk + 1];
        v2f b0, b1;
        b0.x = rw2[kk * 32 + l15];
        b0.y = rw2[(kk + 1) * 32 + l15];
        b1.x = rw2[kk * 32 + 16 + l15];
        b1.y = rw2[(kk + 1) * 32 + 16 + l15];
        acc2 = wmma4(a, b0, acc2);
        acc3 = wmma4(a, b1, acc3);
    }
    float b20 = rb2[l15], b21 = rb2[16 + l15];
#pragma unroll
    for (int rr = 0; rr < 8; ++rr) {
        size_t res = (size_t)(tile * 16 + rr + rowoff);
        res_out[res * 32 + l15]      = gelu_exact(acc2[rr] + b20);
        res_out[res * 32 + 16 + l15] = gelu_exact(acc3[rr] + b21);
    }
}

// ---------------------------------------------------------------------------
// Kernel 4: edge features + edge MLP via f32 WMMA.
// Each wave handles 16 edges (all sharing the same (b,l) since K=32):
//   raw_e[24] = 15 cross-dists | 6 bond-cos | 3 normal-cos, masked by validity
//   edge_emb = gelu(gelu(raw_e @ ew1 + eb1) @ ew2 + eb2)
// ---------------------------------------------------------------------------
__global__ __launch_bounds__(256) void k_edge(
    const float* __restrict__ coords, const float* __restrict__ mask,
    const int* __restrict__ eidx, const float* __restrict__ ws_geo,
    const float* __restrict__ ew1, const float* __restrict__ eb1,
    const float* __restrict__ ew2, const float* __restrict__ eb2,
    float* __restrict__ edge_out)
{
    __shared__ float F[8][16][24];
    __shared__ float Hs[8][16][32];
    int lane = threadIdx.x & 31;
    int wv = threadIdx.x >> 5;
    int tile = blockIdx.x * 8 + wv;   // 65536 tiles total, exact
    int e0 = tile * 16;
    int b = e0 / (LL * KK);
    int rem = e0 - b * (LL * KK);
    int l = rem / KK;
    int k0 = rem - l * KK;            // 0 or 16
    int ri = b * LL + l;

    if (lane < 16) {
        int m = lane;
        int j = eidx[(size_t)ri * KK + k0 + m];
        float validf = ((j >= 0) ? 1.0f : 0.0f) * mask[ri];
        int jc = (j < 0) ? 0 : j;
        int rj = b * LL + jc;
        const float* ci = coords + (size_t)ri * (AT * 3);
        const float* cj = coords + (size_t)rj * (AT * 3);
        float f[24];
        int t = 0;
#pragma unroll
        for (int p = 0; p < AT; ++p)
#pragma unroll
            for (int q = p + 1; q < AT; ++q) {
                float dx = ci[p * 3 + 0] - cj[q * 3 + 0];
                float dy = ci[p * 3 + 1] - cj[q * 3 + 1];
                float dz = ci[p * 3 + 2] - cj[q * 3 + 2];
                f[t++] = sqrtf(dx * dx + dy * dy + dz * dz + EPSF);
            }
        const float* gi = ws_geo + (size_t)ri * 21;
        const float* gj = ws_geo + (size_t)rj * 21;
#pragma unroll
        for (int p = 0; p < 4; ++p)
#pragma unroll
            for (int q = p + 1; q < 4; ++q)
                f[t++] = dot3p(gi + 3 * p, gj + 3 * q);
#pragma unroll
        for (int p = 0; p < 3; ++p)
#pragma unroll
            for (int q = p + 1; q < 3; ++q)
                f[t++] = dot3p(gi + 12 + 3 * p, gj + 12 + 3 * q);
#pragma unroll
        for (int tt = 0; tt < 24; ++tt) F[wv][m][tt] = f[tt] * validf;
    }
    __syncthreads();

    int l15 = lane & 15;
    int koff = (lane < 16) ? 0 : 2;
    int rowoff = (lane < 16) ? 0 : 8;

    v8f acc0 = {0.f, 0.f, 0.f, 0.f, 0.f, 0.f, 0.f, 0.f};
    v8f acc1 = acc0;
#pragma unroll
    for (int c = 0; c < 6; ++c) {           // K = 24
        int kk = 4 * c + koff;
        v2f a;
        a.x = F[wv][l15][kk];
        a.y = F[wv][l15][kk + 1];
        v2f b0, b1;
        b0.x = ew1[kk * 32 + l15];
        b0.y = ew1[(kk + 1) * 32 + l15];
        b1.x = ew1[kk * 32 + 16 + l15];
        b1.y = ew1[(kk + 1) * 32 + 16 + l15];
        acc0 = wmma4(a, b0, acc0);
        acc1 = wmma4(a, b1, acc1);
    }
    float bias0 = eb1[l15], bias1 = eb1[16 + l15];
#pragma unroll
    for (int rr = 0; rr < 8; ++rr) {
        Hs[wv][rr + rowoff][l15]      = gelu_exact(acc0[rr] + bias0);
        Hs[wv][rr + rowoff][16 + l15] = gelu_exact(acc1[rr] + bias1);
    }
    __syncthreads();

    v8f acc2 = {0.f, 0.f, 0.f, 0.f, 0.f, 0.f, 0.f, 0.f};
    v8f acc3 = acc2;
#pragma unroll
    for (int c = 0; c < 8; ++c) {           // K = 32
        int kk = 4 * c + koff;
        v2f a;
        a.x = Hs[wv][l15][kk];
        a.y = Hs[wv][l15][kk + 1];
        v2f b0, b1;
        b0.x = ew2[kk * 32 + l15];
        b0.y = ew2[(kk + 1) * 32 + l15];
        b1.x = ew2[kk * 32 + 16 + l15];
        b1.y = ew2[(kk + 1) * 32 + 16 + l15];
        acc2 = wmma4(a, b0, acc2);
        acc3 = wmma4(a, b1, acc3);
    }
    float b20 = eb2[l15], b21 = eb2[16 + l15];
#pragma unroll
    for (int rr = 0; rr < 8; ++rr) {
        size_t erow = (size_t)(e0 + rr + rowoff);
        edge_out[erow * 32 + l15]      = gelu_exact(acc2[rr] + b20);
        edge_out[erow * 32 + 16 + l15] = gelu_exact(acc3[rr] + b21);
    }
}

// ---------------------------------------------------------------------------
extern "C" void kernel_launch(void* const* d_in, const int* in_sizes, int n_in,
                              void* d_out, int out_size, void* d_ws,
                              size_t ws_size, hipStream_t stream)
{
    const float* coords    = (const float*)d_in[0];
    const float* mask      = (const float*)d_in[1];
    const float* atom_emb  = (const float*)d_in[2];
    const float* atom_mask = (const float*)d_in[3];
    const float* pw1 = (const float*)d_in[4];
    const float* pb1 = (const float*)d_in[5];
    const float* pw2 = (const float*)d_in[6];
    const float* pb2 = (const float*)d_in[7];
    const float* rw1 = (const float*)d_in[8];
    const float* rb1 = (const float*)d_in[9];
    const float* rw2 = (const float*)d_in[10];
    const float* rb2 = (const float*)d_in[11];
    const float* ew1 = (const float*)d_in[12];
    const float* eb1 = (const float*)d_in[13];
    const float* ew2 = (const float*)d_in[14];
    const float* eb2 = (const float*)d_in[15];

    // Workspace layout (~13.1 MB total)
    float* ws_avg = (float*)d_ws;                       // NRES*3
    float* ws_h   = ws_avg + (size_t)NRES * 3;          // NRES*44 (h padded)
    float* ws_geo = ws_h   + (size_t)NRES * 44;         // NRES*21
    int*   ws_eidx = (int*)(ws_geo + (size_t)NRES * 21);// NRES*KK ints

    // Output layout: [res_emb | edge_emb | edge_index(as float)]
    float* out_res  = (float*)d_out;                    // NRES*32
    float* out_edge = out_res  + (size_t)NRES * 32;     // NRES*KK*32
    float* out_eidx = out_edge + (size_t)NRES * KK * 32;// NRES*KK

    k_residue<<<(NRES + 255) / 256, 256, 0, stream>>>(
        coords, mask, atom_emb, atom_mask, pw1, pb1, pw2, pb2,
        ws_avg, ws_h, ws_geo);

    k_topk<<<NRES, 32, 0, stream>>>(ws_avg, mask, ws_eidx, out_eidx);

    k_resmlp<<<NRES / 16 / 8, 256, 0, stream>>>(
        ws_h, rw1, rb1, rw2, rb2, out_res);

    k_edge<<<(NRES * KK) / 16 / 8, 256, 0, stream>>>(
        coords, mask, ws_eidx, ws_geo, ew1, eb1, ew2, eb2, out_edge);
}